// GraphConvolution_31361851196179
// MI455X (gfx1250) — compile-verified
//
#include <hip/hip_runtime.h>

typedef __attribute__((ext_vector_type(2))) float v2f;
typedef __attribute__((ext_vector_type(4))) float v4f;
typedef __attribute__((ext_vector_type(8))) float v8f;

#define IN_F 128
#define OUT_F 32
#define LDS_STRIDE 132          // 128 + 4 pad: fragment reads hit distinct LDS banks
#define GEMM_WAVES 4            // waves per block in the GEMM kernel

// ---------------------------------------------------------------------------
// Kernel 1: support[n][32] = x[n][128] @ W[128][32]   (fp32 WMMA 16x16x4)
// One wave handles a 16-row tile and both 16-wide N halves.
// ---------------------------------------------------------------------------
__global__ __launch_bounds__(GEMM_WAVES * 32)
void gcn_gemm_wmma_kernel(const float* __restrict__ x,
                          const float* __restrict__ w,
                          float* __restrict__ support,
                          int n_nodes) {
    __shared__ float wt[OUT_F * LDS_STRIDE];                // W^T: wt[n][k]
    __shared__ float xs[GEMM_WAVES * 16 * LDS_STRIDE];      // per-wave 16x128 x tile

    const int tid  = threadIdx.x;
    const int lane = tid & 31;
    const int wave = tid >> 5;

    // Stage W^T into LDS (coalesced global reads of row-major W[k][n]).
    for (int i = tid; i < IN_F * OUT_F; i += blockDim.x) {
        const int k = i >> 5;       // row of W
        const int n = i & 31;       // col of W
        wt[n * LDS_STRIDE + k] = w[i];
    }

    const int tile = blockIdx.x * GEMM_WAVES + wave;
    const int row0 = tile * 16;
    // N_NODES (100000) is a multiple of 16 -> every live tile is full.
    const bool active = (row0 + 16) <= n_nodes;
    float* xbase = &xs[wave * 16 * LDS_STRIDE];

    if (active) {
        // Stage this wave's 16x128 x tile: 16 coalesced b128 loads per lane group.
        #pragma unroll
        for (int it = 0; it < 16; ++it) {
            const int f4 = (it * 32 + lane) * 4;   // flat float index, %4 == 0
            const int r  = f4 >> 7;                // /128
            const int c  = f4 & 127;
            const v4f v = *(const v4f*)&x[(row0 + r) * IN_F + c];
            *(v4f*)&xbase[r * LDS_STRIDE + c] = v;
        }
    }
    __syncthreads();            // wt is block-shared; xs is wave-private

    if (!active) return;

    const int m  = lane & 15;   // M row (A, C/D) and N col (B, C/D) for this lane
    const int hi = lane >> 4;   // lane half selects K pair {0,1} vs {2,3}

    v8f c0 = {};                // accumulator, N = 0..15
    v8f c1 = {};                // accumulator, N = 16..31

    #pragma unroll
    for (int k0 = 0; k0 < IN_F; k0 += 4) {
        const int koff = k0 + 2 * hi;
        const v2f a  = *(const v2f*)&xbase[m * LDS_STRIDE + koff];
        const v2f b0 = *(const v2f*)&wt[m * LDS_STRIDE + koff];
        const v2f b1 = *(const v2f*)&wt[(m + 16) * LDS_STRIDE + koff];
        c0 = __builtin_amdgcn_wmma_f32_16x16x4_f32(false, a, false, b0,
                                                   (short)0, c0, false, false);
        c1 = __builtin_amdgcn_wmma_f32_16x16x4_f32(false, a, false, b1,
                                                   (short)0, c1, false, false);
    }

    // C/D layout: VGPR i -> row (i + 8*hi), lanes 0..15 -> N = m, c1 -> N = m+16.
    #pragma unroll
    for (int i = 0; i < 8; ++i) {
        const int r = row0 + i + 8 * hi;
        support[r * OUT_F + m]      = c0[i];
        support[r * OUT_F + m + 16] = c1[i];
    }
}

// ---------------------------------------------------------------------------
// Kernel 2: out[n][j] = bias[j]   (d_out is poisoned by the harness)
// ---------------------------------------------------------------------------
__global__ void gcn_bias_init_kernel(const float* __restrict__ bias,
                                     float* __restrict__ out, int total) {
    const int idx = blockIdx.x * blockDim.x + threadIdx.x;
    if (idx < total) out[idx] = bias[idx & (OUT_F - 1)];
}

// ---------------------------------------------------------------------------
// Kernel 3: out[row[e]] += val[e] * support[col[e]]   (wave per edge,
// lane = feature; coalesced 128B gather + f32 global atomics)
// ---------------------------------------------------------------------------
__global__ __launch_bounds__(256)
void gcn_spmm_kernel(const long long* __restrict__ adj_row,
                     const long long* __restrict__ adj_col,
                     const float* __restrict__ adj_val,
                     const float* __restrict__ support,
                     float* __restrict__ out, int n_edges) {
    const int lane    = threadIdx.x & 31;
    const int wave    = blockIdx.x * (blockDim.x >> 5) + (threadIdx.x >> 5);
    const int n_waves = gridDim.x * (blockDim.x >> 5);

    for (int e = wave; e < n_edges; e += n_waves) {
        const int   r = (int)adj_row[e];
        const int   c = (int)adj_col[e];
        const float v = adj_val[e];
        const float s = support[c * OUT_F + lane];
        atomicAdd(&out[r * OUT_F + lane], v * s);
    }
}

// ---------------------------------------------------------------------------
extern "C" void kernel_launch(void* const* d_in, const int* in_sizes, int n_in,
                              void* d_out, int out_size, void* d_ws, size_t ws_size,
                              hipStream_t stream) {
    const float*     x    = (const float*)d_in[0];
    const long long* arow = (const long long*)d_in[1];
    const long long* acol = (const long long*)d_in[2];
    const float*     aval = (const float*)d_in[3];
    const float*     w    = (const float*)d_in[4];
    const float*     bias = (const float*)d_in[5];
    float* out     = (float*)d_out;
    float* support = (float*)d_ws;              // N_NODES * OUT_F fp32 scratch

    const int n_nodes = in_sizes[0] / IN_F;
    const int n_edges = in_sizes[1];

    const int tiles = (n_nodes + 15) / 16;
    const int gemm_blocks = (tiles + GEMM_WAVES - 1) / GEMM_WAVES;
    gcn_gemm_wmma_kernel<<<gemm_blocks, GEMM_WAVES * 32, 0, stream>>>(x, w, support, n_nodes);

    const int total = n_nodes * OUT_F;
    gcn_bias_init_kernel<<<(total + 255) / 256, 256, 0, stream>>>(bias, out, total);

    gcn_spmm_kernel<<<4096, 256, 0, stream>>>(arow, acol, aval, support, out, n_edges);
}